// STN_86620900426058
// MI455X (gfx1250) — compile-verified
//
#include <hip/hip_runtime.h>
#include <hip/hip_bf16.h>

typedef __attribute__((ext_vector_type(2))) float v2f;
typedef __attribute__((ext_vector_type(8))) float v8f;

#define BATCH    32
#define EMB      128
#define FDIM     64
#define OUTW     512
#define OUTH     512
#define PIX      (OUTW * OUTH)          // 262144
#define BOX_ELEMS  (BATCH * 3 * PIX)    // 25165824
#define GRID_ELEMS (BATCH * PIX * 2)    // 16777216

// ---------------------------------------------------------------------------
// Kernel 1: MLP -> theta, one workgroup, V_WMMA_F32_16X16X4_F32 (full fp32).
//   L1: em1[32,128] = relu(x @ W1^T + b1)   (2x8 tiles, K=128)
//   L2: em2[32, 64] = relu(em1 @ W2^T + b2) (2x4 tiles, K=128)
//   L3: theta[32,6] = (em2 @ Wloc^T + bloc) * rotFix  (2x1 tiles, K=64)
// A-frag (f32 16x4): lane L: M = L%16, K = k0 + 2*(L/16) + {0,1}
// B-frag (f32 4x16): lane L: N = L%16, K = k0 + 2*(L/16) + {0,1}
// D (16x16 f32): VGPR r: M = r + 8*(L/16), N = L%16
// ---------------------------------------------------------------------------
__global__ __launch_bounds__(512) void stn_mlp_wmma(
    const float* __restrict__ x_in,  const float* __restrict__ rotFix,
    const float* __restrict__ W1,    const float* __restrict__ b1,
    const float* __restrict__ W2,    const float* __restrict__ b2,
    const float* __restrict__ Wloc,  const float* __restrict__ bloc,
    float* __restrict__ theta_ws,    float* __restrict__ theta_out)
{
    __shared__ float xs [BATCH * EMB];     // 16 KB
    __shared__ float em1[BATCH * EMB];     // 16 KB
    __shared__ float em2[BATCH * FDIM];    //  8 KB

    const int tid  = threadIdx.x;
    for (int i = tid; i < BATCH * EMB; i += 512) xs[i] = x_in[i];
    __syncthreads();

    const int wave  = tid >> 5;
    const int lane  = tid & 31;
    const int mn    = lane & 15;            // M (for A) and N (for B/D)
    const int khalf = (lane >> 4) << 1;     // 0 or 2
    const int mofs  = (lane >> 4) << 3;     // 0 or 8 (D rows)

    // ---- Layer 1: 16 waves, one 16x16 tile each ---------------------------
    {
        const int Mbase = (wave >> 3) << 4;
        const int Nbase = (wave & 7) << 4;
        const float* arow = &xs[(Mbase + mn) * EMB];
        const float* brow = &W1[(Nbase + mn) * EMB];   // B[k][n] = W1[n][k]
        v8f acc = {};
        for (int k = 0; k < EMB; k += 4) {
            v2f a, b;
            a.x = arow[k + khalf]; a.y = arow[k + khalf + 1];
            b.x = brow[k + khalf]; b.y = brow[k + khalf + 1];
            acc = __builtin_amdgcn_wmma_f32_16x16x4_f32(
                false, a, false, b, (short)0, acc, false, false);
        }
        const float bias = b1[Nbase + mn];
        #pragma unroll
        for (int r = 0; r < 8; ++r) {
            float v = acc[r] + bias;
            em1[(Mbase + mofs + r) * EMB + (Nbase + mn)] = v > 0.f ? v : 0.f;
        }
    }
    __syncthreads();

    // ---- Layer 2: 8 waves -------------------------------------------------
    if (wave < 8) {
        const int Mbase = (wave >> 2) << 4;
        const int Nbase = (wave & 3) << 4;
        const float* arow = &em1[(Mbase + mn) * EMB];
        const float* brow = &W2[(Nbase + mn) * EMB];
        v8f acc = {};
        for (int k = 0; k < EMB; k += 4) {
            v2f a, b;
            a.x = arow[k + khalf]; a.y = arow[k + khalf + 1];
            b.x = brow[k + khalf]; b.y = brow[k + khalf + 1];
            acc = __builtin_amdgcn_wmma_f32_16x16x4_f32(
                false, a, false, b, (short)0, acc, false, false);
        }
        const float bias = b2[Nbase + mn];
        #pragma unroll
        for (int r = 0; r < 8; ++r) {
            float v = acc[r] + bias;
            em2[(Mbase + mofs + r) * FDIM + (Nbase + mn)] = v > 0.f ? v : 0.f;
        }
    }
    __syncthreads();

    // ---- Layer 3: 2 waves (N=6 valid columns of a 16-wide tile) -----------
    if (wave < 2) {
        const int Mbase = wave << 4;
        const float* arow = &em2[(Mbase + mn) * FDIM];
        const bool nvalid = (mn < 6);
        v8f acc = {};
        for (int k = 0; k < FDIM; k += 4) {
            v2f a, b;
            a.x = arow[k + khalf]; a.y = arow[k + khalf + 1];
            b.x = nvalid ? Wloc[mn * FDIM + k + khalf]     : 0.f;
            b.y = nvalid ? Wloc[mn * FDIM + k + khalf + 1] : 0.f;
            acc = __builtin_amdgcn_wmma_f32_16x16x4_f32(
                false, a, false, b, (short)0, acc, false, false);
        }
        if (nvalid) {
            const float bb = bloc[mn];
            #pragma unroll
            for (int r = 0; r < 8; ++r) {
                const int m = Mbase + mofs + r;
                const float t = (acc[r] + bb) * rotFix[m * 6 + mn];
                theta_ws [m * 6 + mn] = t;
                theta_out[m * 6 + mn] = t;
            }
        }
    }
}

// ---------------------------------------------------------------------------
// Kernel 2: affine grid + bilinear sample of the (virtually) 8x-upsampled
// image.  One batch per blockIdx.y; the 48KB source image is staged in LDS so
// all gathers are DS loads; outputs are written with b128 stores (4 pixels per
// thread) since the ~168MB of stores is the roofline cost at 23.3 TB/s.
// ---------------------------------------------------------------------------
__global__ __launch_bounds__(256) void stn_sample(
    const float* __restrict__ newobj, const float* __restrict__ theta,
    float* __restrict__ box, float* __restrict__ grid_out)
{
    __shared__ float img[3 * 64 * 64];      // 48 KB

    const int b = blockIdx.y;
    {   // stage image: 3072 float4s / 256 threads = 12 iters (b128 loads)
        const float4* s4 = (const float4*)(newobj + (size_t)b * (3 * 64 * 64));
        float4* d4 = (float4*)img;
        for (int i = threadIdx.x; i < (3 * 64 * 64) / 4; i += 256) d4[i] = s4[i];
    }
    const float t0 = theta[b * 6 + 0], t1 = theta[b * 6 + 1], t2 = theta[b * 6 + 2];
    const float t3 = theta[b * 6 + 3], t4 = theta[b * 6 + 4], t5 = theta[b * 6 + 5];
    const float dgx = t0 * (1.0f / 256.0f);   // gx step per +1 pixel in x
    const float dgy = t3 * (1.0f / 256.0f);   // gy step per +1 pixel in x
    __syncthreads();

    float* boxb  = box      + (size_t)b * 3 * PIX;
    float* gridb = grid_out + (size_t)b * 2 * PIX;
    const int base = blockIdx.x * (PIX / 64);   // 4096 pixels per block

    #pragma unroll
    for (int it = 0; it < 4; ++it) {
        const int p0 = base + it * 1024 + threadIdx.x * 4;   // 4 px, same row
        const int y = p0 >> 9, x0 = p0 & 511;

        const float xsf = (x0 + 0.5f) * (1.0f / 256.0f) - 1.0f;
        const float ysf = (y  + 0.5f) * (1.0f / 256.0f) - 1.0f;
        float gx = fmaf(t0, xsf, fmaf(t1, ysf, t2));
        float gy = fmaf(t3, xsf, fmaf(t4, ysf, t5));

        float res[3][4];
        float gv[8];

        #pragma unroll
        for (int s = 0; s < 4; ++s) {
            gv[2 * s]     = gx;
            gv[2 * s + 1] = gy;

            // pixel coords in the virtual 512x512 upsampled image
            const float ix = fmaf(gx, 256.f, 255.5f);   // ((gx+1)*512-1)*0.5
            const float iy = fmaf(gy, 256.f, 255.5f);
            const float fx0 = floorf(ix), fy0 = floorf(iy);
            const float wx1 = ix - fx0,   wy1 = iy - fy0;
            const float wx0 = 1.f - wx1,  wy0 = 1.f - wy1;
            const int jx0 = (int)fx0, jy0 = (int)fy0;
            const int jx1 = jx0 + 1,  jy1 = jy0 + 1;

            // zero-padding folded into weights (clamped values are finite)
            const bool vx0 = (unsigned)jx0 < 512u, vx1 = (unsigned)jx1 < 512u;
            const bool vy0 = (unsigned)jy0 < 512u, vy1 = (unsigned)jy1 < 512u;
            const float w00 = (vx0 && vy0) ? wx0 * wy0 : 0.f;
            const float w10 = (vx1 && vy0) ? wx1 * wy0 : 0.f;
            const float w01 = (vx0 && vy1) ? wx0 * wy1 : 0.f;
            const float w11 = (vx1 && vy1) ? wx1 * wy1 : 0.f;

            // nearest-upsample fold: source texel = (j>>3) in the 64x64 image
            const int cx0 = (jx0 < 0 ? 0 : (jx0 > 511 ? 511 : jx0)) >> 3;
            const int cx1 = (jx1 < 0 ? 0 : (jx1 > 511 ? 511 : jx1)) >> 3;
            const int cy0 = (jy0 < 0 ? 0 : (jy0 > 511 ? 511 : jy0)) >> 3;
            const int cy1 = (jy1 < 0 ? 0 : (jy1 > 511 ? 511 : jy1)) >> 3;
            const int i00 = cy0 * 64 + cx0, i10 = cy0 * 64 + cx1;
            const int i01 = cy1 * 64 + cx0, i11 = cy1 * 64 + cx1;

            #pragma unroll
            for (int c = 0; c < 3; ++c) {
                const float* ic = img + c * 4096;
                res[c][s] = ic[i00] * w00 + ic[i10] * w10
                          + ic[i01] * w01 + ic[i11] * w11;
            }
            gx += dgx;
            gy += dgy;
        }

        // wide, aligned stores: 3 x b128 (box) + 2 x b128 (grid)
        #pragma unroll
        for (int c = 0; c < 3; ++c) {
            *(float4*)&boxb[c * PIX + p0] =
                make_float4(res[c][0], res[c][1], res[c][2], res[c][3]);
        }
        float4* g4 = (float4*)&gridb[p0 * 2];
        g4[0] = make_float4(gv[0], gv[1], gv[2], gv[3]);
        g4[1] = make_float4(gv[4], gv[5], gv[6], gv[7]);
    }
}

extern "C" void kernel_launch(void* const* d_in, const int* in_sizes, int n_in,
                              void* d_out, int out_size, void* d_ws, size_t ws_size,
                              hipStream_t stream) {
    const float* newobj = (const float*)d_in[0];
    const float* x_in   = (const float*)d_in[1];
    const float* rotFix = (const float*)d_in[2];
    // d_in[3]/d_in[4] = output_sizex/y (fixed 512, folded into constants)
    const float* W1     = (const float*)d_in[5];
    const float* b1     = (const float*)d_in[6];
    const float* W2     = (const float*)d_in[7];
    const float* b2     = (const float*)d_in[8];
    const float* Wloc   = (const float*)d_in[9];
    const float* bloc   = (const float*)d_in[10];

    float* out       = (float*)d_out;
    float* box       = out;                           // [32,3,512,512]
    float* grid_f    = out + BOX_ELEMS;               // [32,512,512,2]
    float* theta_out = out + BOX_ELEMS + GRID_ELEMS;  // [32,6]
    float* theta_ws  = (float*)d_ws;                  // [32,6] scratch

    stn_mlp_wmma<<<1, 512, 0, stream>>>(x_in, rotFix, W1, b1, W2, b2,
                                        Wloc, bloc, theta_ws, theta_out);

    dim3 grid2(64, BATCH);
    stn_sample<<<grid2, 256, 0, stream>>>(newobj, theta_ws, box, grid_f);
}